// GCNModel_88630945120897
// MI455X (gfx1250) — compile-verified
//
#include <hip/hip_runtime.h>

typedef __attribute__((ext_vector_type(2))) float        v2f;
typedef __attribute__((ext_vector_type(8))) float        v8f;
typedef __attribute__((ext_vector_type(4))) unsigned int v4u;
typedef __attribute__((ext_vector_type(8))) int          v8i;
typedef __attribute__((ext_vector_type(4))) int          v4i;

#define F_DIM 512
#define C_DIM 2
#define KC    64          // K-chunk per TDM tile
#define LDA   65          // 64 dwords + 1 dword TDM pad -> conflict-light LDS stride
#define WAVES 4           // waves (tiles) per block

// ---------------- degree / normalization ----------------

__global__ void gcn_deg_init(float* __restrict__ deg, int N) {
    int i = blockIdx.x * blockDim.x + threadIdx.x;
    if (i < N) deg[i] = 1.0f;  // self-loop weight
}

__global__ void gcn_deg_edges(const int* __restrict__ row,
                              const float* __restrict__ w,
                              float* __restrict__ deg, int E) {
    int e = blockIdx.x * blockDim.x + threadIdx.x;
    if (e < E) atomicAdd(deg + row[e], w[e]);
}

__global__ void gcn_dinv(const float* __restrict__ deg,
                         float* __restrict__ dinv, int N) {
    int i = blockIdx.x * blockDim.x + threadIdx.x;
    if (i < N) {
        float d = deg[i];
        dinv[i] = (d > 0.0f) ? rsqrtf(d) : 0.0f;
    }
}

// ---------------- fused dropout + projection: TDM -> LDS -> WMMA ----------------

__device__ __forceinline__ unsigned hash_u32(unsigned x) {
    x *= 0x9E3779B9u; x ^= x >> 16;
    x *= 0x85EBCA6Bu; x ^= x >> 13;
    x *= 0xC2B2AE35u; x ^= x >> 16;
    return x;
}

__device__ __forceinline__ unsigned rfl(unsigned x) {
    return (unsigned)__builtin_amdgcn_readfirstlane((int)x);
}

// Issue one TDM load of a 16-row x KC-col f32 tile into LDS at lds_off.
// D# per CDNA5 ISA 8.3-8.5: group0 = {count|flags, lds_addr, gaddr_lo, gaddr_hi|type=2}
// group1: data_size=4B, pad_enable, pad_interval=64 dwords, pad_amount=1 dword
//         tensor_dim0=512, tensor_dim1=rows_avail, tile=64x16, dim0_stride=512
__device__ __forceinline__ void tdm_load_tile_16x64(
    const float* gptr, unsigned lds_off, unsigned rows_avail)
{
    unsigned long long ga = (unsigned long long)(uintptr_t)gptr;

    v4u g0;
    g0.x = rfl(1u);                                   // count=1 (valid), user mode
    g0.y = rfl(lds_off);                              // lds_addr (bytes)
    g0.z = rfl((unsigned)ga);                         // global_addr[31:0]
    g0.w = rfl((unsigned)(ga >> 32) | (2u << 30));    // global_addr[56:32] | type=2

    v8i g1;
    g1.s0 = (int)rfl((2u << 16) | (1u << 20) | (5u << 22));  // data_size=4B, pad_en, interval=64dw, amount=1dw
    g1.s1 = (int)rfl(((unsigned)F_DIM & 0xFFFFu) << 16);     // tensor_dim0[15:0]
    g1.s2 = (int)rfl(((unsigned)F_DIM >> 16) |
                     ((rows_avail & 0xFFFFu) << 16));        // tensor_dim0[31:16] | tensor_dim1[15:0]
    g1.s3 = (int)rfl((rows_avail >> 16) | ((unsigned)KC << 16)); // tensor_dim1[31:16] | tile_dim0=64
    g1.s4 = (int)rfl(16u);                                   // tile_dim1=16, tile_dim2=0
    g1.s5 = (int)rfl((unsigned)F_DIM);                       // tensor_dim0_stride = 512
    g1.s6 = 0;                                               // stride hi / dim1_stride lo (unused, 2D)
    g1.s7 = 0;

    v4i z4 = {0, 0, 0, 0};
    v8i z8 = {0, 0, 0, 0, 0, 0, 0, 0};
    __builtin_amdgcn_tensor_load_to_lds(g0, g1, z4, z4, z8, 0);
}

__global__ __launch_bounds__(WAVES * 32) void gcn_dropout_proj(
    const float* __restrict__ x, const float* __restrict__ W,
    float* __restrict__ h, int N)
{
    __shared__ float sW[F_DIM * C_DIM];          // 4 KB, whole W
    __shared__ float sX[WAVES][2][16 * LDA];     // double-buffered TDM tiles, 33.3 KB

    const int tid  = threadIdx.x;
    const int wave = tid >> 5;
    const int lane = tid & 31;

    for (int i = tid; i < F_DIM * C_DIM; i += WAVES * 32) sW[i] = W[i];
    __syncthreads();

    const int row0 = (blockIdx.x * WAVES + wave) * 16;
    if (row0 < N) {                         // wave-uniform: EXEC stays all-1 for WMMA/TDM
        const int m    = lane & 15;         // A row / B col within half-wave
        const int kseg = (lane >> 4) * 2;   // K pair: lanes 0-15 -> {0,1}, 16-31 -> {2,3}
        const unsigned rows_avail = (unsigned)(N - row0);

        // branchless B zero-padding: valid lanes (m < C) read their column,
        // dead lanes read col m&1 (valid address) and multiply by 0
        const int   bcol  = m & 1;
        const float bmask = (m < C_DIM) ? 1.0f : 0.0f;

        const unsigned lds0 = (unsigned)(uintptr_t)(&sX[wave][0][0]);
        const unsigned lds1 = (unsigned)(uintptr_t)(&sX[wave][1][0]);

        v8f acc = {0.f, 0.f, 0.f, 0.f, 0.f, 0.f, 0.f, 0.f};

        // prime the pipeline: chunk 0 -> buffer 0
        tdm_load_tile_16x64(x + (size_t)row0 * F_DIM, lds0, rows_avail);

        for (int k0 = 0; k0 < F_DIM; k0 += KC) {
            const int cur = (k0 / KC) & 1;
            const float* sXc = &sX[wave][cur][0];

            if (k0 + KC < F_DIM) {
                // prefetch next chunk into the other buffer, then wait for current
                tdm_load_tile_16x64(x + (size_t)row0 * F_DIM + (k0 + KC),
                                    cur ? lds0 : lds1, rows_avail);
                __builtin_amdgcn_s_wait_tensorcnt(1);   // in-order: current tile done
            } else {
                __builtin_amdgcn_s_wait_tensorcnt(0);
            }
            asm volatile("" ::: "memory");

            #pragma unroll
            for (int kk = 0; kk < KC; kk += 4) {
                // A 16x4 f32 fragment from LDS; dropout at read time. idx is
                // even (kseg even, kk%4==0), so one hash covers the pair via
                // two independent bits.
                float a0 = sXc[m * LDA + kk + kseg];
                float a1 = sXc[m * LDA + kk + kseg + 1];
                unsigned idx = (unsigned)(row0 + m) * F_DIM +
                               (unsigned)(k0 + kk + kseg);
                unsigned hsh = hash_u32((idx >> 1) ^ 42u);
                a0 = (hsh & 1u) ? a0 * 2.0f : 0.0f;
                a1 = (hsh & 2u) ? a1 * 2.0f : 0.0f;
                v2f a = {a0, a1};

                // B 4x16 f32 fragment, unconditional loads + mask multiply
                int kb = k0 + kk + kseg;
                v2f b;
                b.x = sW[kb * C_DIM + bcol]       * bmask;
                b.y = sW[(kb + 1) * C_DIM + bcol] * bmask;

                acc = __builtin_amdgcn_wmma_f32_16x16x4_f32(
                    false, a, false, b, (short)0, acc, false, false);
            }
            asm volatile("" ::: "memory");
        }

        // D 16x16 f32: VGPR v -> M = v + 8*(lane>=16), N = lane&15 ; keep N < 2
        const int n   = lane & 15;
        const int mhi = (lane >> 4) * 8;
        if (n < C_DIM) {
            #pragma unroll
            for (int v = 0; v < 8; ++v) {
                h[(size_t)(row0 + mhi + v) * C_DIM + n] = acc[v];
            }
        }
    }
}

// ---------------- aggregation ----------------

__global__ void gcn_init_out(const float* __restrict__ h,
                             const float* __restrict__ dinv,
                             const float* __restrict__ b,
                             float* __restrict__ out, int N) {
    int i = blockIdx.x * blockDim.x + threadIdx.x;
    if (i < N) {
        float s = dinv[i] * dinv[i];   // self-loop norm, weight 1
        out[2 * i]     = b[0] + h[2 * i]     * s;
        out[2 * i + 1] = b[1] + h[2 * i + 1] * s;
    }
}

__global__ void gcn_edges(const int* __restrict__ row, const int* __restrict__ col,
                          const float* __restrict__ w, const float* __restrict__ dinv,
                          const float* __restrict__ h, float* __restrict__ out, int E) {
    int e = blockIdx.x * blockDim.x + threadIdx.x;
    if (e >= E) return;
    int r = row[e], c = col[e];
    float nrm = dinv[r] * w[e] * dinv[c];
    float h0 = h[2 * c], h1 = h[2 * c + 1];   // L2-resident gather
    atomicAdd(out + 2 * r,     h0 * nrm);
    atomicAdd(out + 2 * r + 1, h1 * nrm);
}

// ---------------- launch ----------------

extern "C" void kernel_launch(void* const* d_in, const int* in_sizes, int n_in,
                              void* d_out, int out_size, void* d_ws, size_t ws_size,
                              hipStream_t stream) {
    const float* x  = (const float*)d_in[0];
    const int*   ei = (const int*)d_in[1];
    const float* ew = (const float*)d_in[2];
    const float* W  = (const float*)d_in[3];
    const float* b  = (const float*)d_in[4];

    const int N = in_sizes[0] / F_DIM;
    const int E = in_sizes[2];

    const int* row = ei;        // edge_index[0]
    const int* col = ei + E;    // edge_index[1]

    float* out  = (float*)d_out;
    float* deg  = (float*)d_ws;        // N floats
    float* dinv = deg + N;             // N floats
    float* h    = dinv + N;            // 2N floats

    const int T = 256;
    gcn_deg_init <<<(N + T - 1) / T, T, 0, stream>>>(deg, N);
    gcn_deg_edges<<<(E + T - 1) / T, T, 0, stream>>>(row, ew, deg, E);
    gcn_dinv     <<<(N + T - 1) / T, T, 0, stream>>>(deg, dinv, N);

    const int tiles  = (N + 15) / 16;
    const int blocks = (tiles + WAVES - 1) / WAVES;
    gcn_dropout_proj<<<blocks, WAVES * 32, 0, stream>>>(x, W, h, N);

    gcn_init_out<<<(N + T - 1) / T, T, 0, stream>>>(h, dinv, b, out, N);
    gcn_edges   <<<(E + T - 1) / T, T, 0, stream>>>(row, col, ew, dinv, h, out, E);
}